// RMSNorm_39256001085871
// MI455X (gfx1250) — compile-verified
//
#include <hip/hip_runtime.h>
#include <hip/hip_bf16.h>
#include <stdint.h>

#define DIM    3584
#define NOUT   4096
#define MROWS  16384
#define NF4    (DIM / 4)      // 896 float4 per row
#define TPB    128            // 4 waves (wave32)
#define F4PT   (NF4 / TPB)    // 7 float4 per thread per row
#define NBLK   2048           // persistent blocks: 8 rows each

// ---- CDNA5 async-to-LDS path (guarded; falls back to plain loads) ----------
#if defined(__has_builtin)
#  if __has_builtin(__builtin_amdgcn_global_load_async_to_lds_b128)
#    define HAVE_ASYNC_LDS 1
#  endif
#  if __has_builtin(__builtin_amdgcn_s_wait_asynccnt)
#    define HAVE_WAIT_ASYNC 1
#  endif
#endif

typedef int v4i __attribute__((ext_vector_type(4)));
typedef v4i __attribute__((address_space(1)))* gv4i_ptr;  // global int4*
typedef v4i __attribute__((address_space(3)))* lv4i_ptr;  // LDS int4*

__global__ __launch_bounds__(TPB) void rmsnorm_fused_pad_kernel(
    const float* __restrict__ x,
    const float* __restrict__ res,
    const float* __restrict__ w,
    float* __restrict__ out,      // [MROWS, NOUT]
    float* __restrict__ hout)     // [MROWS, DIM]
{
    __shared__ float4 lds_w[NF4]; // 14336 B: weight row staged once per block
    __shared__ float  red[8];     // double-buffered 4-wave partial sums

    const int tid = threadIdx.x;

    // ---- stage weight into LDS (async DMA path on CDNA5) -------------------
#if HAVE_ASYNC_LDS
    {
#pragma unroll
        for (int k = 0; k < F4PT; ++k) {
            const int f = tid + k * TPB;
            __builtin_amdgcn_global_load_async_to_lds_b128(
                (gv4i_ptr)(w + 4 * f),
                (lv4i_ptr)&lds_w[f],
                0, 0);
        }
#if HAVE_WAIT_ASYNC
        __builtin_amdgcn_s_wait_asynccnt(0);
#else
        asm volatile("s_wait_asynccnt 0" ::: "memory");
#endif
    }
#else
    {
        const float4* w4 = (const float4*)w;
#pragma unroll
        for (int k = 0; k < F4PT; ++k) {
            const int f = tid + k * TPB;
            lds_w[f] = w4[f];
        }
    }
#endif
    __syncthreads();

    const int stride = gridDim.x;
    int par = 0;

    for (int row = blockIdx.x; row < MROWS; row += stride, par ^= 1) {
        const float4* xr = (const float4*)(x   + (size_t)row * DIM);
        const float4* rr = (const float4*)(res + (size_t)row * DIM);

        // Prefetch next row's streams (each lane grabs a distinct 128B line).
        if (row + stride < MROWS) {
            const size_t nb = (size_t)(row + stride) * DIM + (size_t)tid * 32;
            __builtin_prefetch(x + nb, 0, 0);
            __builtin_prefetch(res + nb, 0, 0);
        }

        // ---- load, fuse add, accumulate sum of squares ---------------------
        float4 h[F4PT];
        float  ss = 0.0f;
#pragma unroll
        for (int k = 0; k < F4PT; ++k) {
            const int f = tid + k * TPB;
            float4 a = xr[f];
            float4 b = rr[f];
            float4 hh;
            hh.x = a.x + b.x; hh.y = a.y + b.y;
            hh.z = a.z + b.z; hh.w = a.w + b.w;
            h[k] = hh;
            ss += hh.x * hh.x + hh.y * hh.y + hh.z * hh.z + hh.w * hh.w;
        }

        // ---- wave32 reduction + cross-wave LDS reduction -------------------
#pragma unroll
        for (int off = 16; off > 0; off >>= 1)
            ss += __shfl_xor(ss, off, 32);
        if ((tid & 31) == 0)
            red[par * 4 + (tid >> 5)] = ss;
        __syncthreads();

        const float tot = red[par * 4 + 0] + red[par * 4 + 1] +
                          red[par * 4 + 2] + red[par * 4 + 3];
        const float inv = rsqrtf(tot * (1.0f / (float)DIM) + 1e-6f);

        // ---- store h, store scaled out, zero the pad tail ------------------
        float4* h4 = (float4*)(hout + (size_t)row * DIM);
        float4* o4 = (float4*)(out  + (size_t)row * NOUT);
#pragma unroll
        for (int k = 0; k < F4PT; ++k) {
            const int f  = tid + k * TPB;
            float4 hh = h[k];
            h4[f] = hh;
            float4 wv = lds_w[f];
            float4 ov;
            ov.x = hh.x * inv * wv.x;
            ov.y = hh.y * inv * wv.y;
            ov.z = hh.z * inv * wv.z;
            ov.w = hh.w * inv * wv.w;
            o4[f] = ov;
        }
        // pad columns [3584, 4096): 128 threads x float4 = 512 floats
        float4 z; z.x = 0.f; z.y = 0.f; z.z = 0.f; z.w = 0.f;
        ((float4*)(out + (size_t)row * NOUT + DIM))[tid] = z;
    }
}

extern "C" void kernel_launch(void* const* d_in, const int* in_sizes, int n_in,
                              void* d_out, int out_size, void* d_ws, size_t ws_size,
                              hipStream_t stream) {
    (void)in_sizes; (void)n_in; (void)d_ws; (void)ws_size; (void)out_size;
    const float* x   = (const float*)d_in[0];
    const float* res = (const float*)d_in[1];
    const float* w   = (const float*)d_in[2];
    float* out  = (float*)d_out;                       // [16384, 4096]
    float* hout = out + (size_t)MROWS * (size_t)NOUT;  // [16384, 3584]
    rmsnorm_fused_pad_kernel<<<NBLK, TPB, 0, stream>>>(x, res, w, out, hout);
}